// RefSR_60773787239030
// MI455X (gfx1250) — compile-verified
//
#include <hip/hip_runtime.h>
#include <cstdint>

// ---------------------------------------------------------------------------
// USRNet data-step on MI455X (gfx1250).
//   FR  = FkCFy + fft2(alpha*z)
//   FBR = blockmean4x4(Fk*FR); invW = blockmean4x4(F2k)
//   invWBR = FBR / (invW + alpha)
//   FX  = (FR - FkC * tile4x4(invWBR)) / alpha
//   out = real(ifft2(FX))
//
// Bandwidth-bound: 5 fused passes. FFT1024 = radix 16*16*4; the two radix-16
// stages are complex 16x16 matmuls done with V_WMMA_F32_16X16X4_F32, batching
// 16 image lines per workgroup (batch == WMMA N dimension). 16x1024 complex
// tile = 128KB dynamic LDS (MI455X WGP has 320KB).
//
// Tile loads for the three pure-copy prologues use the Tensor Data Mover
// (TENSOR_LOAD_TO_LDS, CDNA5 ISA/8): a 2D D# descriptor
// (group0 = count/lds_addr/global_addr/type, group1 = data_size + dims +
// stride) issued by wave 0, completion via s_wait_tensorcnt + barrier.
//
// Workspace layout (d_ws), total 207,618,048 bytes:
//   [0)            float2 A  : 12M complex  (row-FFT intermediate / IFFT tmp)
//   [100663296)    float2 FR : 12M complex
//   [201326592)    float2 IW : 786432 complex (invWBR, 256x256 per image)
// ---------------------------------------------------------------------------

#define NIMG   12          // N*C = 4*3
#define HW     1024
#define LINES  16
#define TILES_PER_IMG (HW / LINES)   // 64
#define NBLK   (NIMG * TILES_PER_IMG)  // 768
#define SMEM_BYTES (2 * HW * LINES * sizeof(float))  // 131072

typedef float v2f __attribute__((ext_vector_type(2)));
typedef float v8f __attribute__((ext_vector_type(8)));
typedef unsigned int u32x4 __attribute__((ext_vector_type(4)));
typedef int i32x4 __attribute__((ext_vector_type(4)));
typedef int i32x8 __attribute__((ext_vector_type(8)));

#ifndef USE_TDM
#define USE_TDM 1
#endif
#if USE_TDM && defined(__has_builtin)
#if __has_builtin(__builtin_amdgcn_tensor_load_to_lds)
#define HAVE_TDM 1
#endif
#endif
#ifndef HAVE_TDM
#define HAVE_TDM 0
#endif

#if HAVE_TDM
// Issue one 2D TDM load: tile_d1 rows of tile_d0 elements (element size
// 1<<dsize_code bytes), row pitch stride0 elements, into LDS at lds_off.
// Tensor dims == tile dims (tile starts at gaddr), so no OOB clipping.
__device__ __forceinline__ void tdm_load_2d(uint64_t gaddr, uint32_t lds_off,
                                            uint32_t dsize_code,
                                            uint32_t tile_d0, uint32_t tile_d1,
                                            uint32_t stride0) {
  u32x4 g0;
  g0[0] = 1u;                                            // count=1, user mode
  g0[1] = lds_off;                                       // lds_addr (bytes)
  g0[2] = (uint32_t)(gaddr & 0xffffffffull);             // global_addr[31:0]
  g0[3] = (uint32_t)((gaddr >> 32) & 0x01ffffffull)      // global_addr[56:32]
        | (2u << 30);                                    // type = 2 ("image")
  i32x8 g1;
  g1[0] = (int)(dsize_code << 16);                       // data_size; no pad/mcast
  g1[1] = (int)(tile_d0 << 16);                          // tensor_dim0[15:0]
  g1[2] = (int)((tile_d0 >> 16) | (tile_d1 << 16));      // td0 hi | tensor_dim1 lo
  g1[3] = (int)((tile_d1 >> 16) | (tile_d0 << 16));      // td1 hi | tile_dim0
  g1[4] = (int)(tile_d1 & 0xffffu);                      // tile_dim1 (tile_dim2=0)
  g1[5] = (int)stride0;                                  // tensor_dim0_stride lo
  g1[6] = 0;                                             // stride hi / dim1_stride
  g1[7] = 0;
  i32x4 zz = {0, 0, 0, 0};
#if __clang_major__ >= 23
  i32x8 z8 = {0, 0, 0, 0, 0, 0, 0, 0};
  __builtin_amdgcn_tensor_load_to_lds(g0, g1, zz, zz, z8, 0);
#else
  __builtin_amdgcn_tensor_load_to_lds(g0, g1, zz, zz, 0);
#endif
}
#endif  // HAVE_TDM

__device__ __forceinline__ v8f wmma_acc(v2f a, v2f b, v8f c) {
  // D = A(16x4,f32) * B(4x16,f32) + C(16x16,f32)
  return __builtin_amdgcn_wmma_f32_16x16x4_f32(
      /*neg_a=*/false, a, /*neg_b=*/false, b,
      /*c_mod=*/(short)0, c, /*reuse_a=*/false, /*reuse_b=*/false);
}

// One radix-16 butterfly group over 16 batched lines:
//   rows p(r) = pbase + r*pstride  (r = 0..15), columns = 16 lines.
//   D[M][line] = sum_r W16[M][r] * X[r][line], then twiddle exp(i*twmul*M).
// Complex product via 4 real WMMA chains (Ar,Ai precomputed; AiN = -Ai).
__device__ __forceinline__ void dft16_group(
    float* __restrict__ re, float* __restrict__ im,
    int pbase, int pstride, float twmul,
    const v2f* Ar, const v2f* Ai, const v2f* AiN,
    int hi, int col) {
  v2f Br[4], Bi[4];
#pragma unroll
  for (int c = 0; c < 4; ++c) {
    // B 4x16 fragment: vgpr.x holds K=4c+hi, vgpr.y holds K=4c+2+hi, N=col
    int ra = 4 * c + hi;
    int pa = (pbase + ra * pstride) * LINES + col;
    int pb = (pbase + (ra + 2) * pstride) * LINES + col;
    Br[c].x = re[pa]; Br[c].y = re[pb];
    Bi[c].x = im[pa]; Bi[c].y = im[pb];
  }
  v8f Dr = {0.f, 0.f, 0.f, 0.f, 0.f, 0.f, 0.f, 0.f};
  v8f Di = {0.f, 0.f, 0.f, 0.f, 0.f, 0.f, 0.f, 0.f};
#pragma unroll
  for (int c = 0; c < 4; ++c) Dr = wmma_acc(Ar[c], Br[c], Dr);
#pragma unroll
  for (int c = 0; c < 4; ++c) Dr = wmma_acc(AiN[c], Bi[c], Dr);  // -Ai*Bi
#pragma unroll
  for (int c = 0; c < 4; ++c) Di = wmma_acc(Ar[c], Bi[c], Di);
#pragma unroll
  for (int c = 0; c < 4; ++c) Di = wmma_acc(Ai[c], Br[c], Di);
  // twiddle + in-place writeback (same footprint => no intra-stage barrier)
#pragma unroll
  for (int r = 0; r < 8; ++r) {
    int M = r + 8 * hi;          // D layout: vgpr r, lanes 0-15 -> M=r, 16-31 -> M=r+8
    float ts, tc;
    __sincosf(twmul * (float)M, &ts, &tc);
    float dr = Dr[r], di = Di[r];
    int po = (pbase + M * pstride) * LINES + col;
    re[po] = dr * tc - di * ts;
    im[po] = dr * ts + di * tc;
  }
}

// In-place batched 1024-pt complex FFT on a [1024][16] tile in LDS.
// sign = -1 forward, +1 inverse. scale applied at the final un-permute.
// Decomposition: n = 64*n1 + n2 ; k = k1 + 16*j1 + 256*j2
//   stage1: DFT16 over n1 (stride 64), twiddle W1024^(n2*k1)
//   stage2: DFT16 over a  (stride 4),  twiddle W64^(b*j1)      (n2 = 4a+b)
//   stage3: radix-4 over b (VALU), then digit-reversal fixup to natural order.
__device__ __forceinline__ void fft1024_tile(
    float* __restrict__ re, float* __restrict__ im,
    float sign, float scale, int t) {
  const float TWO_PI = 6.2831853071795864769f;
  int l = t & 31;
  int wave = t >> 5;          // 8 waves of 32 (wave32)
  int hi = l >> 4;
  int col = l & 15;

  // A fragments: W16[m][k] = exp(i*sign*2pi*m*k/16); A 16x4 layout per ISA:
  // lanes 0-15: M=lane, vgpr.x=K0 of chunk, .y=K1 ; lanes 16-31: K2,K3.
  v2f Ar[4], Ai[4], AiN[4];
#pragma unroll
  for (int c = 0; c < 4; ++c) {
    int k0 = 4 * c + 2 * hi;
    float s0, c0, s1, c1;
    __sincosf(sign * TWO_PI * (float)(col * k0) * (1.f / 16.f), &s0, &c0);
    __sincosf(sign * TWO_PI * (float)(col * (k0 + 1)) * (1.f / 16.f), &s1, &c1);
    Ar[c].x = c0;  Ar[c].y = c1;
    Ai[c].x = s0;  Ai[c].y = s1;
    AiN[c].x = -s0; AiN[c].y = -s1;
  }

  __syncthreads();

  // stage 1: 64 groups (n2), 8 per wave; rows p = 64*r + n2
#pragma unroll 1
  for (int gi = 0; gi < 8; ++gi) {
    int n2 = wave * 8 + gi;
    dft16_group(re, im, n2, 64,
                sign * TWO_PI * (float)n2 * (1.f / 1024.f),
                Ar, Ai, AiN, hi, col);
  }
  __syncthreads();

  // stage 2: 64 groups (k1,b); rows p = 64*k1 + 4*a + b
#pragma unroll 1
  for (int gi = 0; gi < 8; ++gi) {
    int g = wave * 8 + gi;
    int k1 = g >> 2, b = g & 3;
    dft16_group(re, im, 64 * k1 + b, 4,
                sign * TWO_PI * (float)b * (1.f / 64.f),
                Ar, Ai, AiN, hi, col);
  }
  __syncthreads();

  // stage 3: radix-4 over b (w4 = sign*i), positions p = 64*k1 + 4*j1 + b
  {
    int line = t & 15;
    int ib = t >> 4;
#pragma unroll 1
    for (int i = 0; i < 16; ++i) {
      int idx = ib + (i << 4);           // 0..255
      int k1 = idx & 15, j1 = idx >> 4;
      int p0 = (64 * k1 + 4 * j1) * LINES + line;
      float r0 = re[p0],      i0 = im[p0];
      float r1 = re[p0 + 16], i1 = im[p0 + 16];
      float r2 = re[p0 + 32], i2 = im[p0 + 32];
      float r3 = re[p0 + 48], i3 = im[p0 + 48];
      float t0r = r0 + r2, t0i = i0 + i2;
      float t1r = r0 - r2, t1i = i0 - i2;
      float t2r = r1 + r3, t2i = i1 + i3;
      float t3r = r1 - r3, t3i = i1 - i3;
      float u3r = -sign * t3i, u3i = sign * t3r;   // (sign*i)*t3
      re[p0]      = t0r + t2r; im[p0]      = t0i + t2i;
      re[p0 + 16] = t1r + u3r; im[p0 + 16] = t1i + u3i;
      re[p0 + 32] = t0r - t2r; im[p0 + 32] = t0i - t2i;
      re[p0 + 48] = t1r - u3r; im[p0 + 48] = t1i - u3i;
    }
  }
  __syncthreads();

  // un-permute digit-reversed result to natural order, apply scale.
  // storage p(k) = 64*(k&15) + 4*((k>>4)&15) + (k>>8)
  {
    int line = t & 15;
    int kb = t >> 4;
    float rr[64], ri[64];
#pragma unroll
    for (int i = 0; i < 64; ++i) {
      int k = kb + (i << 4);
      int p = ((k & 15) << 6) + (((k >> 4) & 15) << 2) + (k >> 8);
      rr[i] = re[p * LINES + line] * scale;
      ri[i] = im[p * LINES + line] * scale;
    }
    __syncthreads();
#pragma unroll
    for (int i = 0; i < 64; ++i) {
      int k = kb + (i << 4);
      re[k * LINES + line] = rr[i];
      im[k * LINES + line] = ri[i];
    }
  }
  __syncthreads();
}

// ---------------------------------------------------------------------------
// Pass 1: rows forward: A[nc][h][:] = fft1024_w(alpha[n] * z[nc][h][:])
__global__ void k_row_fwd(const float* __restrict__ z,
                          const float* __restrict__ alpha,
                          float2* __restrict__ A) {
  extern __shared__ float sm[];
  float* re = sm;
  float* im = sm + HW * LINES;
  int t = threadIdx.x;
  int blk = blockIdx.x;
  int nc = blk >> 6;
  int h0 = (blk & 63) << 4;
  float a = alpha[nc / 3];

  int line = t >> 4;           // row within tile
  int s = t & 15;
#if HAVE_TDM
  // TDM: 16 rows x 1024 f32, row-major into the im plane (staging), then a
  // register round-trip transpose into the [point][line] re/im planes.
  if (t < 32) {
    uint64_t ga = (uint64_t)(uintptr_t)z +
                  (((uint64_t)nc << 20) + (uint64_t)h0 * HW) * 4ull;
    tdm_load_2d(ga, (uint32_t)(uintptr_t)im, /*dsize=*/2u, HW, LINES, HW);
    __builtin_amdgcn_s_wait_tensorcnt(0);
  }
  __syncthreads();
  {
    float vals[64];
#pragma unroll
    for (int i = 0; i < 64; ++i) vals[i] = im[line * HW + (s + (i << 4))];
    __syncthreads();
#pragma unroll
    for (int i = 0; i < 64; ++i) {
      int p = s + (i << 4);
      re[p * LINES + line] = a * vals[i];
      im[p * LINES + line] = 0.f;
    }
  }
#else
  const float* rowp = z + ((size_t)nc << 20) + (size_t)(h0 + line) * HW;
#pragma unroll 4
  for (int i = 0; i < 64; ++i) {
    int p = s + (i << 4);
    if (i + 16 < 64) __builtin_prefetch(rowp + p + 256, 0, 1);
    re[p * LINES + line] = a * rowp[p];
    im[p * LINES + line] = 0.f;
  }
#endif

  fft1024_tile(re, im, -1.f, 1.f, t);

  float2* op = A + ((size_t)nc << 20) + (size_t)h0 * HW + (size_t)line * HW;
#pragma unroll 4
  for (int i = 0; i < 64; ++i) {
    int k = s + (i << 4);
    float2 v; v.x = re[k * LINES + line]; v.y = im[k * LINES + line];
    op[k] = v;
  }
}

// Pass 2: columns forward + FkCFy:  FR[:, w] = FkCFy[:, w] + fft1024_h(A[:, w])
__global__ void k_col_fwd(const float2* __restrict__ A,
                          const float2* __restrict__ FkCFy,
                          float2* __restrict__ FR) {
  extern __shared__ float sm[];
  float* re = sm;
  float* im = sm + HW * LINES;
  int t = threadIdx.x;
  int blk = blockIdx.x;
  int nc = blk >> 6;
  int w0 = (blk & 63) << 4;

  int line = t & 15;           // column (w) within tile -> fast in lanes
  int s = t >> 4;
  size_t base = ((size_t)nc << 20) + (size_t)(w0 + line);
#if HAVE_TDM
  // TDM: 1024 rows x 16 complex (128B rows, pitch 8KB) -> interleaved
  // [h][line] float2 staging over the whole 128KB, then split to re/im.
  if (t < 32) {
    uint64_t ga = (uint64_t)(uintptr_t)A +
                  (((uint64_t)nc << 20) + (uint64_t)w0) * 8ull;
    tdm_load_2d(ga, (uint32_t)(uintptr_t)sm, /*dsize=*/3u, LINES, HW, HW);
    __builtin_amdgcn_s_wait_tensorcnt(0);
  }
  __syncthreads();
  {
    const float2* stage = (const float2*)sm;
    float2 vals[64];
#pragma unroll
    for (int i = 0; i < 64; ++i) vals[i] = stage[(s + (i << 4)) * LINES + line];
    __syncthreads();
#pragma unroll
    for (int i = 0; i < 64; ++i) {
      int h = s + (i << 4);
      re[h * LINES + line] = vals[i].x;
      im[h * LINES + line] = vals[i].y;
    }
  }
#else
#pragma unroll 4
  for (int i = 0; i < 64; ++i) {
    int h = s + (i << 4);
    if (i + 16 < 64) __builtin_prefetch(A + base + ((size_t)(h + 256) * HW), 0, 1);
    float2 v = A[base + (size_t)h * HW];
    re[h * LINES + line] = v.x;
    im[h * LINES + line] = v.y;
  }
#endif

  fft1024_tile(re, im, -1.f, 1.f, t);

#pragma unroll 4
  for (int i = 0; i < 64; ++i) {
    int h = s + (i << 4);
    float2 add = FkCFy[base + (size_t)h * HW];
    float2 v;
    v.x = re[h * LINES + line] + add.x;
    v.y = im[h * LINES + line] + add.y;
    FR[base + (size_t)h * HW] = v;
  }
}

// Pass 3: invWBR = blockmean(Fk*FR) / (blockmean(F2k) + alpha)
__global__ void k_reduce(const float2* __restrict__ Fk,
                         const float2* __restrict__ FR,
                         const float2* __restrict__ F2k,
                         const float* __restrict__ alpha,
                         float2* __restrict__ IW) {
  int idx = blockIdx.x * 256 + threadIdx.x;     // 0 .. 12*256*256-1
  int j = idx & 255;
  int i = (idx >> 8) & 255;
  int nc = idx >> 16;
  float al = alpha[nc / 3];

  float br = 0.f, bi = 0.f, wr = 0.f, wi = 0.f;
#pragma unroll
  for (int p = 0; p < 4; ++p) {
#pragma unroll
    for (int q = 0; q < 4; ++q) {
      size_t off = ((size_t)nc << 20) + (size_t)(p * 256 + i) * HW + (size_t)(q * 256 + j);
      float2 fk = Fk[off];
      float2 fr = FR[off];
      float2 f2 = F2k[off];
      br += fk.x * fr.x - fk.y * fr.y;
      bi += fk.x * fr.y + fk.y * fr.x;
      wr += f2.x;
      wi += f2.y;
    }
  }
  br *= (1.f / 16.f); bi *= (1.f / 16.f);
  wr = wr * (1.f / 16.f) + al;              // alpha adds to real part
  wi *= (1.f / 16.f);
  float den = 1.f / (wr * wr + wi * wi);
  float2 o;
  o.x = (br * wr + bi * wi) * den;
  o.y = (bi * wr - br * wi) * den;
  IW[((size_t)nc << 16) + (size_t)(i << 8) + j] = o;
}

// Pass 4: FX = (FR - FkC*tile(invWBR))/alpha, then inverse column FFT -> A
__global__ void k_fx_colinv(const float2* __restrict__ FR,
                            const float2* __restrict__ FkC,
                            const float2* __restrict__ IW,
                            const float* __restrict__ alpha,
                            float2* __restrict__ A) {
  extern __shared__ float sm[];
  float* re = sm;
  float* im = sm + HW * LINES;
  int t = threadIdx.x;
  int blk = blockIdx.x;
  int nc = blk >> 6;
  int w0 = (blk & 63) << 4;
  float inv_al = 1.f / alpha[nc / 3];

  int line = t & 15;
  int s = t >> 4;
  int wj = (w0 + line) & 255;
  size_t base = ((size_t)nc << 20) + (size_t)(w0 + line);
  size_t iwb = ((size_t)nc << 16) + (size_t)wj;
#pragma unroll 4
  for (int i = 0; i < 64; ++i) {
    int h = s + (i << 4);
    float2 fr = FR[base + (size_t)h * HW];
    float2 fc = FkC[base + (size_t)h * HW];
    float2 iw = IW[iwb + (size_t)((h & 255) << 8)];
    float pr = fc.x * iw.x - fc.y * iw.y;
    float pi = fc.x * iw.y + fc.y * iw.x;
    re[h * LINES + line] = (fr.x - pr) * inv_al;
    im[h * LINES + line] = (fr.y - pi) * inv_al;
  }

  fft1024_tile(re, im, +1.f, 1.f / 1024.f, t);   // half of the 1/(H*W)

#pragma unroll 4
  for (int i = 0; i < 64; ++i) {
    int h = s + (i << 4);
    float2 v;
    v.x = re[h * LINES + line];
    v.y = im[h * LINES + line];
    A[base + (size_t)h * HW] = v;
  }
}

// Pass 5: inverse row FFT, take real part -> out
__global__ void k_row_inv(const float2* __restrict__ A,
                          float* __restrict__ out) {
  extern __shared__ float sm[];
  float* re = sm;
  float* im = sm + HW * LINES;
  int t = threadIdx.x;
  int blk = blockIdx.x;
  int nc = blk >> 6;
  int h0 = (blk & 63) << 4;

  int line = t >> 4;
  int s = t & 15;
#if HAVE_TDM
  // TDM: 16 rows x 1024 complex, row-major interleaved [line][p] float2 over
  // the whole 128KB, then register round-trip split into re/im planes.
  if (t < 32) {
    uint64_t ga = (uint64_t)(uintptr_t)A +
                  (((uint64_t)nc << 20) + (uint64_t)h0 * HW) * 8ull;
    tdm_load_2d(ga, (uint32_t)(uintptr_t)sm, /*dsize=*/3u, HW, LINES, HW);
    __builtin_amdgcn_s_wait_tensorcnt(0);
  }
  __syncthreads();
  {
    const float2* stage = (const float2*)sm;
    float2 vals[64];
#pragma unroll
    for (int i = 0; i < 64; ++i) vals[i] = stage[line * HW + (s + (i << 4))];
    __syncthreads();
#pragma unroll
    for (int i = 0; i < 64; ++i) {
      int p = s + (i << 4);
      re[p * LINES + line] = vals[i].x;
      im[p * LINES + line] = vals[i].y;
    }
  }
#else
  const float2* rowp = A + ((size_t)nc << 20) + (size_t)(h0 + line) * HW;
#pragma unroll 4
  for (int i = 0; i < 64; ++i) {
    int p = s + (i << 4);
    if (i + 16 < 64) __builtin_prefetch(rowp + p + 256, 0, 1);
    float2 v = rowp[p];
    re[p * LINES + line] = v.x;
    im[p * LINES + line] = v.y;
  }
#endif

  fft1024_tile(re, im, +1.f, 1.f / 1024.f, t);   // other half of 1/(H*W)

  float* op = out + ((size_t)nc << 20) + (size_t)(h0 + line) * HW;
#pragma unroll 4
  for (int i = 0; i < 64; ++i) {
    int k = s + (i << 4);
    op[k] = re[k * LINES + line];
  }
}

// ---------------------------------------------------------------------------
extern "C" void kernel_launch(void* const* d_in, const int* in_sizes, int n_in,
                              void* d_out, int out_size, void* d_ws, size_t ws_size,
                              hipStream_t stream) {
  const float*  z     = (const float*)d_in[0];
  const float2* Fk    = (const float2*)d_in[1];
  const float2* FkC   = (const float2*)d_in[2];
  const float2* F2k   = (const float2*)d_in[3];
  const float2* FkCFy = (const float2*)d_in[4];
  const float*  alpha = (const float*)d_in[5];
  // d_in[6] = sf (==4, hardcoded)

  char* ws = (char*)d_ws;
  float2* A  = (float2*)(ws);                       // 100,663,296 B
  float2* FR = (float2*)(ws + 100663296ull);        // 100,663,296 B
  float2* IW = (float2*)(ws + 201326592ull);        //   6,291,456 B

  float* outp = (float*)d_out;

  k_row_fwd  <<<NBLK, 256, SMEM_BYTES, stream>>>(z, alpha, A);
  k_col_fwd  <<<NBLK, 256, SMEM_BYTES, stream>>>(A, FkCFy, FR);
  k_reduce   <<<(NIMG * 256 * 256) / 256, 256, 0, stream>>>(Fk, FR, F2k, alpha, IW);
  k_fx_colinv<<<NBLK, 256, SMEM_BYTES, stream>>>(FR, FkC, IW, alpha, A);
  k_row_inv  <<<NBLK, 256, SMEM_BYTES, stream>>>(A, outp);
}